// GQA_62191126446621
// MI455X (gfx1250) — compile-verified
//
#include <hip/hip_runtime.h>
#include <hip/hip_bf16.h>

// ---------------------------------------------------------------------------
// GQA decode (B=32, SQ=1, DIM=4096, NH=32, NKV=8, HD=128, kv_len=4096), fp32.
// Bandwidth-bound (~1.16 GB @ 23.3 TB/s); fp32 WMMA 16x16x4 for all matmuls.
// v2: compile-time GEMM strides (kills 64-bit address mul chains) and
//     K-split-across-8-waves GEMM with LDS reduction for 8x occupancy.
// ---------------------------------------------------------------------------

typedef float v2f __attribute__((ext_vector_type(2)));
typedef float v8f __attribute__((ext_vector_type(8)));

#define GB    32
#define GDIM  4096
#define GNH   32
#define GNKV  8
#define GHD   128
#define GNREP 4
#define GSEQ  4096
#define GLAST 4095

__device__ __forceinline__ v8f wmma4f32(v2f a, v2f b, v8f c) {
  // (neg_a, A, neg_b, B, c_mod, C, reuse_a, reuse_b)
  return __builtin_amdgcn_wmma_f32_16x16x4_f32(false, a, false, b,
                                               (short)0, c, false, false);
}

// ---------------------------------------------------------------------------
// fp32 GEMM: C[M,N] = A[M,K] @ B[K,N], row-major. N, K compile-time.
// block = 256 (8 waves), grid = (N/16, M/16). The 8 waves split K into 8
// ranges; each computes a partial 16x16 tile with V_WMMA_F32_16X16X4_F32,
// partials reduced through LDS.
// ---------------------------------------------------------------------------
template <int N, int K>
__global__ __launch_bounds__(256) void gemm_f32_wmma(
    const float* __restrict__ A, const float* __restrict__ B,
    float* __restrict__ C)
{
  __shared__ float part[8 * 256];      // 8 partial 16x16 tiles

  const int lane  = threadIdx.x & 31;
  const int wave  = threadIdx.x >> 5;
  const int half  = lane >> 4;         // 0: K0/K1 lanes, 1: K2/K3 lanes
  const int idx16 = lane & 15;         // M-row (A) / N-col (B)
  const int n0 = blockIdx.x * 16;
  const int m0 = blockIdx.y * 16;

  constexpr int KW = K / 8;            // K-range per wave
  const int k0 = wave * KW;

  const float* arow = A + (size_t)(m0 + idx16) * K + k0 + half * 2;
  const float* bcol = B + (size_t)(k0 + half * 2) * N + n0 + idx16;

  v8f acc = {};
#pragma unroll 8
  for (int k = 0; k < KW; k += 4) {
    v2f a = *reinterpret_cast<const v2f*>(arow + k);     // A[m][k0+k+2h], +1
    v2f b;
    b.x = bcol[(size_t)k * N];                           // B[k0+k+2h][n]
    b.y = bcol[(size_t)(k + 1) * N];                     // B[k0+k+2h+1][n]
    acc = wmma4f32(a, b, acc);
  }

  // stash partial tile: slot layout keeps the C register mapping
  float* my = part + wave * 256 + lane;
#pragma unroll
  for (int r = 0; r < 8; ++r)
    my[r * 32] = acc[r];
  __syncthreads();

  // reduce 8 partials and store: t = r*32 + lane
  {
    const int t  = threadIdx.x;
    const int l  = t & 31;
    const int r  = t >> 5;
    float s = 0.f;
#pragma unroll
    for (int w = 0; w < 8; ++w)
      s += part[w * 256 + t];
    const int row = r + 8 * (l >> 4);
    const int col = l & 15;
    C[(size_t)(m0 + row) * N + n0 + col] = s;
  }
}

// ---------------------------------------------------------------------------
// RoPE (interleaved pairs) applied in-place to ws copies of xq and xk.
// ---------------------------------------------------------------------------
__global__ __launch_bounds__(256) void rope_kernel(
    float* __restrict__ xq, float* __restrict__ xk,
    const float* __restrict__ cosv, const float* __restrict__ sinv)
{
  const int qpairs = GB * GNH * (GHD / 2);
  const int kpairs = GB * GNKV * (GHD / 2);
  int idx = blockIdx.x * 256 + threadIdx.x;
  float* buf;
  int pair;
  if (idx < qpairs)               { buf = xq; pair = idx; }
  else if (idx < qpairs + kpairs) { buf = xk; pair = idx - qpairs; }
  else return;
  int i = pair & 63;
  float c = cosv[i], s = sinv[i];
  size_t base = (size_t)pair * 2;
  float t0 = buf[base], t1 = buf[base + 1];
  buf[base]     = t0 * c - t1 * s;
  buf[base + 1] = t0 * s + t1 * c;
}

// ---------------------------------------------------------------------------
// Attention: one block per (b, kv_head). 256 threads = 8 waves.
// Phase 1: scores (q padded 4->16 rows) via WMMA, NT-stream K once -> LDS.
// Phase 2: softmax over 4096 per rep.
// Phase 3: probs @ V via WMMA, NT-stream V once; position 4095 uses fresh k/v.
// ---------------------------------------------------------------------------
__global__ __launch_bounds__(256) void attn_kernel(
    const float* __restrict__ xq,      // (B, NH, HD) roped
    const float* __restrict__ xk_new,  // (B, NKV, HD) roped
    const float* __restrict__ xv_new,  // (B, NKV, HD)
    const float* __restrict__ cache_k, // (B, SEQ, NKV, HD)
    const float* __restrict__ cache_v, // (B, SEQ, NKV, HD)
    float* __restrict__ attn_out)      // (B, NH, HD)
{
  __shared__ float sc[GNREP * GSEQ];   // 64 KB: scores -> probs
  __shared__ float red[16];            // [0..7] wave scratch, [8] gmax, [12..15] 1/sum

  const int b    = blockIdx.x >> 3;
  const int h    = blockIdx.x & 7;
  const int lane  = threadIdx.x & 31;
  const int wave  = threadIdx.x >> 5;
  const int half  = lane >> 4;
  const int idx16 = lane & 15;
  const int rep4  = idx16 & 3;         // clamp padded rows to valid addresses

  constexpr int kvstride = GNKV * GHD; // floats per seq position (constant)

  // ------------------- Phase 1: scores = q @ K^T / sqrt(HD) -------------------
  const float* qbase = xq + ((size_t)b * GNH + h * GNREP + rep4) * GHD + half * 2;
  for (int tile = wave; tile < GSEQ / 16; tile += 8) {
    const int p0  = tile * 16;
    const int pos = p0 + idx16;
    const float* krow = (pos == GLAST)
        ? (xk_new + ((size_t)b * GNKV + h) * GHD)
        : (cache_k + ((size_t)b * GSEQ + pos) * kvstride + h * GHD);
    const float* k2 = krow + half * 2;
    v8f acc = {};
#pragma unroll 8
    for (int kk = 0; kk < GHD; kk += 4) {
      v2f a = *reinterpret_cast<const v2f*>(qbase + kk);
      v2f bb = __builtin_nontemporal_load(reinterpret_cast<const v2f*>(k2 + kk));
      acc = wmma4f32(a, bb, acc);
    }
    if (half == 0) {
      const float scale = 0.08838834764831845f;  // 1/sqrt(128)
#pragma unroll
      for (int r = 0; r < GNREP; ++r)
        sc[r * GSEQ + p0 + idx16] = acc[r] * scale;
    }
  }
  __syncthreads();

  // ------------------- Phase 2: softmax over seq per rep ----------------------
  for (int r = 0; r < GNREP; ++r) {
    float m = -1e30f;
    for (int i = threadIdx.x; i < GSEQ; i += 256)
      m = fmaxf(m, sc[r * GSEQ + i]);
    for (int off = 16; off >= 1; off >>= 1)
      m = fmaxf(m, __shfl_xor(m, off, 32));
    if (lane == 0) red[wave] = m;
    __syncthreads();
    if (threadIdx.x == 0) {
      float g = red[0];
      for (int w = 1; w < 8; ++w) g = fmaxf(g, red[w]);
      red[8] = g;
    }
    __syncthreads();
    const float gmax = red[8];
    float s = 0.f;
    for (int i = threadIdx.x; i < GSEQ; i += 256) {
      float p = __expf(sc[r * GSEQ + i] - gmax);
      sc[r * GSEQ + i] = p;
      s += p;
    }
    for (int off = 16; off >= 1; off >>= 1)
      s += __shfl_xor(s, off, 32);
    if (lane == 0) red[wave] = s;
    __syncthreads();
    if (threadIdx.x == 0) {
      float g = 0.f;
      for (int w = 0; w < 8; ++w) g += red[w];
      red[12 + r] = 1.0f / g;
    }
    __syncthreads();
  }

  // ------------------- Phase 3: out = probs @ V -------------------------------
  const int d0 = wave * 16;            // each wave owns 16 of the 128 dims
  const float* vdim = cache_v + (size_t)b * GSEQ * kvstride + h * GHD + d0 + idx16;
  const float* prow = sc + (size_t)rep4 * GSEQ + half * 2;

  v8f acc = {};
#pragma unroll 8
  for (int t = 0; t < GSEQ - 4; t += 4) {
    v2f a = *reinterpret_cast<const v2f*>(prow + t);
    const int kp = t + half * 2;
    v2f bb;
    bb.x = __builtin_nontemporal_load(vdim + (size_t)kp * kvstride);
    bb.y = __builtin_nontemporal_load(vdim + (size_t)(kp + 1) * kvstride);
    acc = wmma4f32(a, bb, acc);
  }
  {  // epilogue tile t = 4092: position 4095 comes from fresh v
    const int t  = GSEQ - 4;
    v2f a = *reinterpret_cast<const v2f*>(prow + t);
    const int kp = t + half * 2;        // 4092 or 4094
    v2f bb;
    bb.x = vdim[(size_t)kp * kvstride];
    const int kp1 = kp + 1;             // 4093 or 4095
    float vy = vdim[(size_t)kp1 * kvstride];   // pos 4095 row exists (stale)
    if (kp1 == GLAST)
      vy = xv_new[((size_t)b * GNKV + h) * GHD + d0 + idx16];
    bb.y = vy;
    acc = wmma4f32(a, bb, acc);
  }
  if (half == 0) {
#pragma unroll
    for (int r = 0; r < GNREP; ++r) {
      const float inv = red[12 + r];
      attn_out[((size_t)b * GNH + h * GNREP + r) * GHD + d0 + idx16] = acc[r] * inv;
    }
  }
}

// ---------------------------------------------------------------------------
// Launcher
// ---------------------------------------------------------------------------
extern "C" void kernel_launch(void* const* d_in, const int* in_sizes, int n_in,
                              void* d_out, int out_size, void* d_ws, size_t ws_size,
                              hipStream_t stream) {
  (void)in_sizes; (void)n_in; (void)out_size; (void)ws_size;
  const float* x       = (const float*)d_in[0];
  const float* cache_k = (const float*)d_in[1];
  const float* cache_v = (const float*)d_in[2];
  const float* fcos    = (const float*)d_in[3];
  const float* fsin    = (const float*)d_in[4];
  const float* wq      = (const float*)d_in[5];
  const float* wk      = (const float*)d_in[6];
  const float* wv      = (const float*)d_in[7];
  const float* wo      = (const float*)d_in[8];
  float* out = (float*)d_out;

  float* ws  = (float*)d_ws;
  float* xq   = ws;                       // 32*32*128  = 131072
  float* xk   = xq + GB * GNH * GHD;      // 32*8*128   =  32768
  float* xv   = xk + GB * GNKV * GHD;     // 32*8*128   =  32768
  float* attn = xv + GB * GNKV * GHD;     // 32*32*128  = 131072

  // xq = x @ wq  (32 x 4096 x 4096): 256x2 blocks, 8 waves each
  gemm_f32_wmma<GNH * GHD, GDIM>
      <<<dim3(GNH * GHD / 16, GB / 16), 256, 0, stream>>>(x, wq, xq);
  // xk = x @ wk ; xv = x @ wv  (32 x 1024 x 4096)
  gemm_f32_wmma<GNKV * GHD, GDIM>
      <<<dim3(GNKV * GHD / 16, GB / 16), 256, 0, stream>>>(x, wk, xk);
  gemm_f32_wmma<GNKV * GHD, GDIM>
      <<<dim3(GNKV * GHD / 16, GB / 16), 256, 0, stream>>>(x, wv, xv);
  // RoPE on xq, xk
  {
    int pairs = GB * GNH * (GHD / 2) + GB * GNKV * (GHD / 2);   // 81920
    rope_kernel<<<(pairs + 255) / 256, 256, 0, stream>>>(xq, xk, fcos, fsin);
  }
  // attention: one block per (b, kv_head)
  attn_kernel<<<GB * GNKV, 256, 0, stream>>>(xq, xk, xv, cache_k, cache_v, attn);
  // out = attn @ wo  (32 x 4096 x 4096)
  gemm_f32_wmma<GDIM, GDIM>
      <<<dim3(GDIM / 16, GB / 16), 256, 0, stream>>>(attn, wo, out);
}